// GIN_43293270343767
// MI455X (gfx1250) — compile-verified
//
#include <hip/hip_runtime.h>
#include <hip/hip_bf16.h>

typedef __attribute__((ext_vector_type(2))) float v2f;
typedef __attribute__((ext_vector_type(8))) float v8f;

#define DCH   128
#define SW_LD 132   // padded LDS row stride (floats) to avoid bank conflicts

// ---------------- utility: zero a float buffer (float4 granularity) ----------------
__global__ void zero_f32(float* __restrict__ p, long long n4) {
    long long i = (long long)blockIdx.x * blockDim.x + threadIdx.x;
    if (i < n4) ((float4*)p)[i] = make_float4(0.f, 0.f, 0.f, 0.f);
}

// ---------------- in-degree count ----------------
__global__ void deg_count(const int* __restrict__ dst, float* __restrict__ deg, int nE) {
    int i = blockIdx.x * blockDim.x + threadIdx.x;
    if (i < nE) atomicAdd(deg + dst[i], 1.0f);
}

__global__ void deg_invert(const float* __restrict__ deg, float* __restrict__ dinv, int n) {
    int i = blockIdx.x * blockDim.x + threadIdx.x;
    if (i < n) {
        float d = deg[i];
        dinv[i] = (d > 0.f) ? (1.0f / d) : 0.f;
    }
}

// ---------------- edge scatter: agg[dst] += x[src] * edge_w ----------------
// 32 threads per edge, each handling a float4 chunk of the 128-dim feature row.
__global__ void gin_scatter(const float* __restrict__ x, const float* __restrict__ ew,
                            const int* __restrict__ src, const int* __restrict__ dst,
                            float* __restrict__ agg, int nE) {
    long long t = (long long)blockIdx.x * blockDim.x + threadIdx.x;
    int e = (int)(t >> 5);
    if (e >= nE) return;
    int c = ((int)t & 31) * 4;
    float w = ew[e];
    const float4 v = *(const float4*)(x + (size_t)src[e] * DCH + c);
    float* a = agg + (size_t)dst[e] * DCH + c;
    atomicAdd(a + 0, v.x * w);
    atomicAdd(a + 1, v.y * w);
    atomicAdd(a + 2, v.z * w);
    atomicAdd(a + 3, v.w * w);
}

// ---------------- fused combine + GEMM via f32 WMMA ----------------
// out[n, j] = act( sum_k ((1+eps)*x[n,k] + agg[n,k]*dinv[n]) * W[j,k] + b[j] )
// One wave -> 16x128 output strip (8 16x16 tiles). K=128 reduced with 32x V_WMMA_F32_16X16X4_F32.
__global__ __launch_bounds__(256) void gin_gemm_wmma(
    const float* __restrict__ x, const float* __restrict__ agg,
    const float* __restrict__ dinv, const float* __restrict__ eps_arr, int eps_idx,
    const float* __restrict__ W, const float* __restrict__ bias,
    float* __restrict__ out, int n_nodes, int do_relu, int zero_row0)
{
    extern __shared__ float sW[];  // [128][SW_LD]
    const int tid = threadIdx.x;

    // cooperative load of W (row-major 128x128) into padded LDS
    const float4* W4 = (const float4*)W;
    for (int i = tid; i < DCH * DCH / 4; i += 256) {
        int row = i >> 5;      // 32 float4 per row
        int q   = i & 31;
        *(float4*)(&sW[row * SW_LD + q * 4]) = W4[i];
    }
    __syncthreads();

    const float eps = eps_arr[eps_idx];
    const float s   = 1.0f + eps;
    const int wave = tid >> 5;
    const int lane = tid & 31;
    const int half = lane >> 4;   // lane group 0..15 vs 16..31
    const int l16  = lane & 15;
    const int row0 = blockIdx.x * 128 + wave * 16;
    int node = row0 + l16;
    if (node >= n_nodes) node = n_nodes - 1;   // clamp loads; stores are guarded
    const float di = dinv[node];
    const float* xr = x   + (size_t)node * DCH;
    const float* ar = agg + (size_t)node * DCH;

    // Build A fragments for the whole K=128 once (fused GIN combine).
    // A (16x4 f32): lanes 0-15 hold K={4kb+0,4kb+1}, lanes 16-31 hold K={4kb+2,4kb+3}.
    v2f A[32];
#pragma unroll
    for (int kb = 0; kb < 32; ++kb) {
        const int k = kb * 4 + half * 2;
        float2 xv = *(const float2*)(xr + k);
        float2 av = *(const float2*)(ar + k);
        A[kb].x = s * xv.x + av.x * di;
        A[kb].y = s * xv.y + av.y * di;
    }

#pragma unroll 1
    for (int j = 0; j < 8; ++j) {
        const int col = j * 16 + l16;
        const float* wrow = &sW[col * SW_LD];
        v8f c = {};
#pragma unroll
        for (int kb = 0; kb < 32; ++kb) {
            const int k = kb * 4 + half * 2;
            v2f b;
            b.x = wrow[k];       // B (4x16): VGPR0 = K {0,2}, VGPR1 = K {1,3}
            b.y = wrow[k + 1];
            c = __builtin_amdgcn_wmma_f32_16x16x4_f32(
                    /*neg_a=*/false, A[kb], /*neg_b=*/false, b,
                    /*c_mod=*/(short)0, c, /*reuse_a=*/false, /*reuse_b=*/false);
        }
        const float bv = bias[col];
#pragma unroll
        for (int r = 0; r < 8; ++r) {
            const int rr = row0 + half * 8 + r;   // C/D: VGPR r -> M = r (+8 for hi half)
            if (rr < n_nodes) {
                float v = c[r] + bv;
                if (do_relu) v = fmaxf(v, 0.0f);
                if (zero_row0 && rr == 0) v = 0.0f;
                out[(size_t)rr * DCH + col] = v;
            }
        }
    }
}

extern "C" void kernel_launch(void* const* d_in, const int* in_sizes, int n_in,
                              void* d_out, int out_size, void* d_ws, size_t ws_size,
                              hipStream_t stream) {
    const float* emb = (const float*)d_in[0];
    const float* w1  = (const float*)d_in[1];
    const float* b1  = (const float*)d_in[2];
    const float* w2  = (const float*)d_in[3];
    const float* b2  = (const float*)d_in[4];
    const float* eps = (const float*)d_in[5];
    const float* ew  = (const float*)d_in[6];
    const int*   src = (const int*)d_in[7];
    const int*   dst = (const int*)d_in[8];

    const int N = in_sizes[0] / DCH;
    const int E = in_sizes[6];

    // workspace layout: deg[N] | dinv[N] | agg[N*128] | bufA[N*128]
    float* deg  = (float*)d_ws;
    float* dinv = deg + N;
    float* agg  = dinv + N;
    float* bufA = agg + (size_t)N * DCH;
    float* out  = (float*)d_out;

    dim3 b256(256);
    auto zero = [&](float* p, size_t nf) {
        long long n4 = (long long)(nf / 4);
        zero_f32<<<(unsigned)((n4 + 255) / 256), b256, 0, stream>>>(p, n4);
    };

    // ---- degree (once; graph topology shared across layers) ----
    zero(deg, (size_t)N);
    deg_count<<<(E + 255) / 256, b256, 0, stream>>>(dst, deg, E);
    deg_invert<<<(N + 255) / 256, b256, 0, stream>>>(deg, dinv, N);

    const int    gemmBlocks    = (N + 127) / 128;
    const size_t shmem         = (size_t)DCH * SW_LD * sizeof(float);
    const long long scatThreads = (long long)E * 32;
    const unsigned scatBlocks  = (unsigned)((scatThreads + 255) / 256);

    // ---- layer 1: emb -> bufA (relu) ----
    zero(agg, (size_t)N * DCH);
    gin_scatter<<<scatBlocks, b256, 0, stream>>>(emb, ew, src, dst, agg, E);
    gin_gemm_wmma<<<gemmBlocks, b256, shmem, stream>>>(emb, agg, dinv, eps, 0,
                                                       w1, b1, bufA, N, 1, 0);

    // ---- layer 2: bufA -> bufA in place (each block reads only rows it writes) ----
    zero(agg, (size_t)N * DCH);
    gin_scatter<<<scatBlocks, b256, 0, stream>>>(bufA, ew, src, dst, agg, E);
    gin_gemm_wmma<<<gemmBlocks, b256, shmem, stream>>>(bufA, agg, dinv, eps, 1,
                                                       w2, b2, bufA, N, 1, 0);

    // ---- layer 3: bufA -> d_out (no relu, row 0 zeroed) ----
    zero(agg, (size_t)N * DCH);
    gin_scatter<<<scatBlocks, b256, 0, stream>>>(bufA, ew, src, dst, agg, E);
    gin_gemm_wmma<<<gemmBlocks, b256, shmem, stream>>>(bufA, agg, dinv, eps, 2,
                                                       w2, b2, out, N, 0, 1);
}